// SPPNet_90254442758174
// MI455X (gfx1250) — compile-verified
//
#include <hip/hip_runtime.h>
#include <hip/hip_bf16.h>

typedef float v4f __attribute__((ext_vector_type(4)));

#define NWG     2048          // workgroups in stage 1
#define NT      256           // threads per workgroup (8 wave32 waves)
#define PLANES  16384         // 32 batch * 512 channels
#define PF4     1024          // float4 per 64x64 plane (4096 floats)
#define PPW     (PLANES/NWG)  // 8 planes per workgroup
#define NSLOT   14

#if defined(__gfx1250__)
// CDNA5 async path: global -> LDS bulk copy, tracked by ASYNCcnt.
// SADDR form: address = SGPR64 base + 32-bit per-lane VGPR offset + imm offset.
__device__ __forceinline__ void async_b128(unsigned lds_byte_off, unsigned voff,
                                           const float* sbase, int ioff) {
  asm volatile("global_load_async_to_lds_b128 %0, %1, %2 offset:%3"
               :: "v"(lds_byte_off), "v"(voff), "s"(sbase), "n"(ioff)
               : "memory");
}
#define WAIT_ASYNCCNT(n) asm volatile("s_wait_asynccnt %0" :: "n"(n) : "memory")
#endif

__device__ __forceinline__ v4f vmax4(v4f a, v4f b) {
  v4f r;
  r.x = fmaxf(a.x, b.x);
  r.y = fmaxf(a.y, b.y);
  r.z = fmaxf(a.z, b.z);
  r.w = fmaxf(a.w, b.w);
  return r;
}

// Stage 1: stream all planes, per-WG partial max for the 14 SPP slots.
__global__ __launch_bounds__(NT) void spp_stage1(const float* __restrict__ feat,
                                                 float* __restrict__ partials) {
  const int t  = threadIdx.x;
  const int wg = blockIdx.x;
  const float NEG = -__builtin_inff();

  v4f acc0 = {NEG, NEG, NEG, NEG};
  v4f acc1 = acc0, acc2 = acc0, acc3 = acc0;

#if defined(__gfx1250__)
  // 3 x 16KB stream buffers, aliased with the 14KB epilogue reduce array
  // (disjoint phases, separated by a barrier).
  __shared__ union {
    v4f   buf[3][PF4];
    float red[NT * NSLOT];
  } sh;
  float* red = sh.red;
  {
    const unsigned voff = (unsigned)t * 16u;        // per-lane byte offset in plane
    const unsigned lb[3] = {(unsigned)(size_t)&sh.buf[0][t],
                            (unsigned)(size_t)&sh.buf[1][t],
                            (unsigned)(size_t)&sh.buf[2][t]};

#define ISSUE_PLANE(lbase, plane) do {                         \
      const float* pb = feat + (size_t)(plane) * 4096;         \
      async_b128((lbase) +     0u, voff, pb,     0);           \
      async_b128((lbase) +  4096u, voff, pb,  4096);           \
      async_b128((lbase) +  8192u, voff, pb,  8192);           \
      async_b128((lbase) + 12288u, voff, pb, 12288);           \
    } while (0)

    // prologue: 2 planes in flight
    ISSUE_PLANE(lb[0], wg);
    ISSUE_PLANE(lb[1], wg + NWG);

#pragma unroll
    for (int i = 0; i < PPW; ++i) {
      if (i + 2 < PPW) {
        ISSUE_PLANE(lb[(i + 2) % 3], wg + (i + 2) * NWG);
        WAIT_ASYNCCNT(8);                 // plane i retired (in-order completion)
      } else if (i + 1 < PPW) {
        WAIT_ASYNCCNT(4);
      } else {
        WAIT_ASYNCCNT(0);                 // drain
      }
      const v4f* b = sh.buf[i % 3];
      acc0 = vmax4(acc0, b[t + 0]);
      acc1 = vmax4(acc1, b[t + 256]);
      acc2 = vmax4(acc2, b[t + 512]);
      acc3 = vmax4(acc3, b[t + 768]);
    }
#undef ISSUE_PLANE
  }
  __syncthreads();   // all waves done streaming before red[] aliases buf[]
#else
  // Fallback (host pass / non-gfx1250): direct nontemporal streaming loads.
  __shared__ float red_arr[NT * NSLOT];
  float* red = red_arr;
  const v4f* gbase = (const v4f*)feat;
  for (int i = 0; i < PPW; ++i) {
    const v4f* g = gbase + (size_t)(wg + i * NWG) * PF4;
    acc0 = vmax4(acc0, __builtin_nontemporal_load(g + t + 0));
    acc1 = vmax4(acc1, __builtin_nontemporal_load(g + t + 256));
    acc2 = vmax4(acc2, __builtin_nontemporal_load(g + t + 512));
    acc3 = vmax4(acc3, __builtin_nontemporal_load(g + t + 768));
  }
#endif

  // Map the 16 owned spatial positions to the 14 SPP slots, reduce in LDS.
#pragma unroll
  for (int s = 0; s < NSLOT; ++s) red[t * NSLOT + s] = NEG;

  float vals[16] = {acc0.x, acc0.y, acc0.z, acc0.w,
                    acc1.x, acc1.y, acc1.z, acc1.w,
                    acc2.x, acc2.y, acc2.z, acc2.w,
                    acc3.x, acc3.y, acc3.z, acc3.w};
#pragma unroll
  for (int j = 0; j < 4; ++j) {
#pragma unroll
    for (int e = 0; e < 4; ++e) {
      const int p = (t + 256 * j) * 4 + e;   // flat spatial index 0..4095
      const int h = p >> 6;
      const int w = p & 63;
      const float v = vals[j * 4 + e];
      // k=1 (slot 0)
      red[t * NSLOT + 0] = fmaxf(red[t * NSLOT + 0], v);
      // k=2: 32x32 quadrants (slots 1..4)
      const int s1 = 1 + (h >> 5) * 2 + (w >> 5);
      red[t * NSLOT + s1] = fmaxf(red[t * NSLOT + s1], v);
      // k=3: 21x21 blocks covering [0,63) (slots 5..13)
      if (h < 63 && w < 63) {
        const int s2 = 5 + (h / 21) * 3 + (w / 21);
        red[t * NSLOT + s2] = fmaxf(red[t * NSLOT + s2], v);
      }
    }
  }
  __syncthreads();
  for (int off = NT / 2; off >= 1; off >>= 1) {
    if (t < off) {
#pragma unroll
      for (int s = 0; s < NSLOT; ++s)
        red[t * NSLOT + s] = fmaxf(red[t * NSLOT + s], red[(t + off) * NSLOT + s]);
    }
    __syncthreads();
  }
  if (t < NSLOT) partials[(size_t)wg * NSLOT + t] = red[t];
}

// Stage 2: reduce NWG x 14 partials -> 14 outputs.
__global__ __launch_bounds__(NT) void spp_stage2(const float* __restrict__ partials,
                                                 float* __restrict__ out) {
  const int t = threadIdx.x;
  const float NEG = -__builtin_inff();
  float loc[NSLOT];
#pragma unroll
  for (int s = 0; s < NSLOT; ++s) loc[s] = NEG;

  for (int r = t; r < NWG; r += NT) {
    const float* row = partials + (size_t)r * NSLOT;
#pragma unroll
    for (int s = 0; s < NSLOT; ++s) loc[s] = fmaxf(loc[s], row[s]);
  }

  __shared__ float red[NT * NSLOT];
#pragma unroll
  for (int s = 0; s < NSLOT; ++s) red[t * NSLOT + s] = loc[s];
  __syncthreads();
  for (int off = NT / 2; off >= 1; off >>= 1) {
    if (t < off) {
#pragma unroll
      for (int s = 0; s < NSLOT; ++s)
        red[t * NSLOT + s] = fmaxf(red[t * NSLOT + s], red[(t + off) * NSLOT + s]);
    }
    __syncthreads();
  }
  if (t < NSLOT) out[t] = red[t];
}

extern "C" void kernel_launch(void* const* d_in, const int* in_sizes, int n_in,
                              void* d_out, int out_size, void* d_ws, size_t ws_size,
                              hipStream_t stream) {
  (void)in_sizes; (void)n_in; (void)out_size; (void)ws_size;
  const float* feat = (const float*)d_in[0];
  float* out = (float*)d_out;
  float* partials = (float*)d_ws;  // needs NWG*14*4 = 112 KB scratch

  spp_stage1<<<NWG, NT, 0, stream>>>(feat, partials);
  spp_stage2<<<1, NT, 0, stream>>>(partials, out);
}